// IDWT_Transform_43722767073763
// MI455X (gfx1250) — compile-verified
//
#include <hip/hip_runtime.h>
#include <cstdint>

// Haar inverse DWT2, fp32, shapes fixed by the reference:
// B=16, C=64, H2=128, W2=128  ->  out (16,64,256,256)
// Memory-bound streaming kernel: 256 MiB in + 256 MiB out @ 23.3 TB/s ~ 22us.
// Uses CDNA5 async global->LDS loads (ASYNCcnt) double-buffered per wave,
// wave32 128-bit accesses, and nontemporal stores.

typedef float f4 __attribute__((ext_vector_type(4)));

#define WAVES   8            // waves per block (256 threads, wave32)
#define ROWF    128          // W2: coefficients per subband row
#define ROWB    (ROWF * 4)   // bytes per subband row (512)

// Async copy of 16 bytes per lane from global memory into LDS.
// vdst = per-lane LDS byte offset, vaddr = per-lane 64-bit global address.
__device__ __forceinline__ void async_ld16(uint32_t lds_off, const void* gaddr) {
  asm volatile("global_load_async_to_lds_b128 %0, %1, off"
               :: "v"(lds_off), "v"(gaddr)
               : "memory");
}
__device__ __forceinline__ void wait_async_le4() {
  asm volatile("s_wait_asynccnt 0x4" ::: "memory");
}
__device__ __forceinline__ void wait_async_0() {
  asm volatile("s_wait_asynccnt 0x0" ::: "memory");
}

__global__ __launch_bounds__(WAVES * 32) void haar_idwt2_async(
    const float* __restrict__ LL, const float* __restrict__ LH,
    const float* __restrict__ HL, const float* __restrict__ HH,
    float* __restrict__ out, int nrows) {
  // Per-wave private staging: 2 stages x 4 subbands x 128 floats = 4 KB/wave.
  __shared__ __align__(16) float smem[WAVES * 2 * 4 * ROWF];

  const int wave = threadIdx.x >> 5;   // wave32 on gfx1250
  const int lane = threadIdx.x & 31;

  float* wbuf = &smem[wave * (2 * 4 * ROWF)];

  auto issue = [&](int r, int stage) {
    const uint64_t gofs = ((uint64_t)r * ROWF + (uint64_t)(lane << 2)) * sizeof(float);
    // Low 32 bits of a generic LDS pointer == workgroup-relative LDS byte offset.
    const uint32_t lofs =
        (uint32_t)(uintptr_t)(void*)(wbuf + stage * (4 * ROWF)) + (uint32_t)(lane << 4);
    async_ld16(lofs + 0 * ROWB, (const char*)LL + gofs);
    async_ld16(lofs + 1 * ROWB, (const char*)LH + gofs);
    async_ld16(lofs + 2 * ROWB, (const char*)HL + gofs);
    async_ld16(lofs + 3 * ROWB, (const char*)HH + gofs);
  };

  const int rstride = gridDim.x * WAVES;
  int r = blockIdx.x * WAVES + wave;
  if (r < nrows) issue(r, 0);
  int stage = 0;

  for (; r < nrows; r += rstride) {
    const int rn = r + rstride;
    if (rn < nrows) {
      issue(rn, stage ^ 1);   // 8 async loads in flight
      wait_async_le4();       // current row's 4 have landed in LDS
    } else {
      wait_async_0();
    }

    const f4* buf = (const f4*)(wbuf + stage * (4 * ROWF));
    const f4 ll = buf[0 * 32 + lane];
    const f4 lh = buf[1 * 32 + lane];
    const f4 hl = buf[2 * 32 + lane];
    const f4 hh = buf[3 * 32 + lane];

    // a=(LL+LH+HL+HH)/2  b=(LL+LH-HL-HH)/2  c=(LL-LH+HL-HH)/2  d=(LL-LH-HL+HH)/2
    const f4 s0 = ll + lh;
    const f4 s1 = hl + hh;
    const f4 d0 = ll - lh;
    const f4 d1 = hl - hh;
    const f4 va = (s0 + s1) * 0.5f;
    const f4 vb = (s0 - s1) * 0.5f;
    const f4 vc = (d0 + d1) * 0.5f;
    const f4 vd = (d0 - d1) * 0.5f;

    // r = plane*128 + i ; output plane is 256x256, rows 2i and 2i+1.
    const int plane = r >> 7;
    const int i     = r & 127;
    float* o0 = out + (uint64_t)plane * (2 * ROWF) * (2 * ROWF)
                    + (uint64_t)(2 * i) * (2 * ROWF) + (uint64_t)(lane << 3);
    float* o1 = o0 + 2 * ROWF;

    const f4 r0a = {va.x, vb.x, va.y, vb.y};
    const f4 r0b = {va.z, vb.z, va.w, vb.w};
    const f4 r1a = {vc.x, vd.x, vc.y, vd.y};
    const f4 r1b = {vc.z, vd.z, vc.w, vd.w};
    __builtin_nontemporal_store(r0a, (f4*)o0);
    __builtin_nontemporal_store(r0b, (f4*)(o0 + 4));
    __builtin_nontemporal_store(r1a, (f4*)o1);
    __builtin_nontemporal_store(r1b, (f4*)(o1 + 4));

    stage ^= 1;
  }
}

extern "C" void kernel_launch(void* const* d_in, const int* in_sizes, int n_in,
                              void* d_out, int out_size, void* d_ws, size_t ws_size,
                              hipStream_t stream) {
  (void)n_in; (void)out_size; (void)d_ws; (void)ws_size;
  const float* LL = (const float*)d_in[0];
  const float* LH = (const float*)d_in[1];
  const float* HL = (const float*)d_in[2];
  const float* HH = (const float*)d_in[3];
  float* out = (float*)d_out;

  const int nrows = in_sizes[0] / ROWF;        // B*C*H2 = 131072 coefficient rows
  const int grid  = 4096;                      // 32768 waves, 4 rows each
  haar_idwt2_async<<<grid, WAVES * 32, 0, stream>>>(LL, LH, HL, HH, out, nrows);
}